// GraphAttentionQKV_66262755442760
// MI455X (gfx1250) — compile-verified
//
#include <hip/hip_runtime.h>

#define LEAKF 0.2f

typedef float v2f __attribute__((ext_vector_type(2)));
typedef float v8f __attribute__((ext_vector_type(8)));

#define Bsz 4
#define Nsz 1024
#define Dsz 64
#define Hsz 64

// ---- gfx1250 async global->LDS copy (ASYNCcnt path), with safe fallback ----
#if defined(__has_builtin)
#if __has_builtin(__builtin_amdgcn_global_load_async_to_lds_b128)
#define HAVE_ASYNC_LDS 1
#endif
#endif
#ifndef HAVE_ASYNC_LDS
#define HAVE_ASYNC_LDS 0
#endif

#if HAVE_ASYNC_LDS
typedef int async_v4i __attribute__((vector_size(16)));
typedef __attribute__((address_space(1))) async_v4i* async_gptr;
typedef __attribute__((address_space(3))) async_v4i* async_lptr;
#define ASYNC_COPY_B128(gptr, lptr)                                         \
    __builtin_amdgcn_global_load_async_to_lds_b128(                         \
        (async_gptr)(gptr), (async_lptr)(lptr), 0, 0)
#if __has_builtin(__builtin_amdgcn_s_wait_asynccnt)
#define WAIT_ASYNC() __builtin_amdgcn_s_wait_asynccnt(0)
#else
#define WAIT_ASYNC() asm volatile("s_wait_asynccnt 0x0" ::: "memory")
#endif
#endif

// ---------------- Kernel 1: fused projections ----------------
// Hq[row][h]  = sum_d q[row][d] * (Wq[d][h] + Wd[d][h])
// Hkb[row][h] = bias[h] + sum_d k[row][d] * (Wk[d][h] - Wd[d][h])
__global__ __launch_bounds__(256) void proj_kernel(
    const float* __restrict__ qs, const float* __restrict__ ks,
    const float* __restrict__ W, const float* __restrict__ bias,
    float* __restrict__ Hq, float* __restrict__ Hkb)
{
    __shared__ float sWqd[64 * 64];
    __shared__ float sWkd[64 * 64];
    __shared__ float sB[64];
    int tid = threadIdx.x;
    for (int f = tid; f < 4096; f += 256) {
        float wq = W[f];
        float wk = W[4096 + f];
        float wd = W[8192 + f];
        sWqd[f] = wq + wd;
        sWkd[f] = wk - wd;
    }
    if (tid < 64) sB[tid] = bias[tid];
    __syncthreads();

    int row = blockIdx.x * 4 + (tid >> 6);   // 4096 rows total
    int h   = tid & 63;
    const float* q = qs + row * 64;
    const float* k = ks + row * 64;
    float aq = 0.f;
    float ak = sB[h];
#pragma unroll 8
    for (int d = 0; d < 64; ++d) {
        aq = fmaf(q[d], sWqd[d * 64 + h], aq);
        ak = fmaf(k[d], sWkd[d * 64 + h], ak);
    }
    Hq[row * 64 + h]  = aq;
    Hkb[row * 64 + h] = ak;
}

// ---------------- Kernel 2: scores + softmax + P@V (WMMA) ----------------
#define SS_STRIDE 1028   // 1024 + 4 pad -> bank = (4*i + j) % 64, conflict free, 16B aligned
#define HK_STRIDE 68     // 64 + 4 pad

__global__ __launch_bounds__(256) void attn_kernel(
    const float* __restrict__ Hq, const float* __restrict__ Hkb,
    const float* __restrict__ vs, const float* __restrict__ avec,
    float* __restrict__ out)
{
    __shared__ float sS[16 * SS_STRIDE];   // score / P tile, 16 queries x 1024 keys
    __shared__ float sHk[64 * HK_STRIDE];  // staged Hkb chunk
    __shared__ float sA[64];
    __shared__ float sOp[4 * 256];         // K-half partial accumulators
    __shared__ float sLinv[16];            // 1/rowsum

    int tid = threadIdx.x;
    int b   = blockIdx.x >> 6;             // 64 query-tiles per batch
    int i0  = (blockIdx.x & 63) << 4;

    if (tid < 16) ((float4*)sA)[tid] = ((const float4*)avec)[tid];

    // ---- Phase A: scores ----
    int iq = tid & 15;                     // query row of this thread
    int jg = tid >> 4;                     // key group of this thread
    const float* hqrow = Hq + (size_t)(b * Nsz + i0 + iq) * 64;
    float4 hq[16];
#pragma unroll
    for (int t = 0; t < 16; ++t) hq[t] = ((const float4*)hqrow)[t];

    const float* hkbase = Hkb + (size_t)b * Nsz * 64;

    for (int jc = 0; jc < Nsz; jc += 64) {
        __syncthreads();
        // stage 64 rows of Hkb (+bias) into LDS
#if HAVE_ASYNC_LDS
#pragma unroll
        for (int rep = 0; rep < 4; ++rep) {
            int f  = rep * 1024 + tid * 4;
            int rr = f >> 6;
            int cc = f & 63;
            float* gsrc = (float*)&hkbase[(size_t)(jc + rr) * 64 + cc];
            ASYNC_COPY_B128(gsrc, &sHk[rr * HK_STRIDE + cc]);
        }
        WAIT_ASYNC();
#else
#pragma unroll
        for (int rep = 0; rep < 4; ++rep) {
            int f  = rep * 1024 + tid * 4;
            int rr = f >> 6;
            int cc = f & 63;
            float4 v = *(const float4*)&hkbase[(size_t)(jc + rr) * 64 + cc];
            *(float4*)&sHk[rr * HK_STRIDE + cc] = v;
        }
#endif
        __syncthreads();
#pragma unroll
        for (int jj = 0; jj < 4; ++jj) {
            int jloc = jg * 4 + jj;
            const float* hk = &sHk[jloc * HK_STRIDE];
            float acc = 0.f;
#pragma unroll
            for (int h4 = 0; h4 < 16; ++h4) {
                float4 k4 = ((const float4*)hk)[h4];
                float4 a4 = ((const float4*)sA)[h4];
                float4 q4 = hq[h4];
                float t;
                t = q4.x + k4.x; t = fmaxf(t, LEAKF * t); acc = fmaf(a4.x, t, acc);
                t = q4.y + k4.y; t = fmaxf(t, LEAKF * t); acc = fmaf(a4.y, t, acc);
                t = q4.z + k4.z; t = fmaxf(t, LEAKF * t); acc = fmaf(a4.z, t, acc);
                t = q4.w + k4.w; t = fmaxf(t, LEAKF * t); acc = fmaf(a4.w, t, acc);
            }
            sS[iq * SS_STRIDE + jc + jloc] = 0.125f * acc;  // * 1/sqrt(D)
        }
    }
    __syncthreads();

    // ---- Phase B: softmax per row (16 lanes per row, half-wave shuffles) ----
    {
        int r = tid >> 4;
        int t = tid & 15;
        float* row = &sS[r * SS_STRIDE];
        float mx = -3.4e38f;
#pragma unroll 8
        for (int c = 0; c < 64; ++c) mx = fmaxf(mx, row[t + 16 * c]);
        mx = fmaxf(mx, __shfl_xor(mx, 8, 16));
        mx = fmaxf(mx, __shfl_xor(mx, 4, 16));
        mx = fmaxf(mx, __shfl_xor(mx, 2, 16));
        mx = fmaxf(mx, __shfl_xor(mx, 1, 16));
        float sum = 0.f;
#pragma unroll 8
        for (int c = 0; c < 64; ++c) {
            int idx = t + 16 * c;
            float p = __expf(row[idx] - mx);
            row[idx] = p;
            sum += p;
        }
        sum += __shfl_xor(sum, 8, 16);
        sum += __shfl_xor(sum, 4, 16);
        sum += __shfl_xor(sum, 2, 16);
        sum += __shfl_xor(sum, 1, 16);
        if (t == 0) sLinv[r] = 1.0f / sum;
    }
    __syncthreads();

    // ---- Phase C: O = P @ V with V_WMMA_F32_16X16X4_F32 ----
    // 8 waves: ntile = wave&3 (which 16 output cols), khalf = wave>>2 (K split)
    int lane  = tid & 31;
    int wave  = tid >> 5;
    int ntile = wave & 3;
    int khalf = wave >> 2;
    int m     = lane & 15;            // A-matrix row for this lane
    int ksel  = (lane >> 4) << 1;     // upper half-wave carries K+2, K+3
    int n     = (ntile << 4) + (lane & 15);
    const float* vb = vs + (size_t)b * Nsz * 64 + n;

    v8f acc = {};
    int kbase = khalf * 512;
#pragma unroll 4
    for (int kk = 0; kk < 512; kk += 4) {
        int k = kbase + kk;
        // A: P[m][k+ksel], P[m][k+ksel+1]  (8B-aligned LDS float2)
        const float* ap = &sS[m * SS_STRIDE + k + ksel];
        v2f av;
        av.x = ap[0];
        av.y = ap[1];
        // B: V[k+ksel][n], V[k+ksel+1][n]
        const float* bp = vb + (size_t)(k + ksel) * 64;
        v2f bv;
        bv.x = bp[0];
        bv.y = bp[64];
        acc = __builtin_amdgcn_wmma_f32_16x16x4_f32(false, av, false, bv,
                                                    (short)0, acc, false, false);
    }

    if (khalf == 1) {
#pragma unroll
        for (int r = 0; r < 8; ++r) sOp[ntile * 256 + r * 32 + lane] = acc[r];
    }
    __syncthreads();
    if (khalf == 0) {
#pragma unroll
        for (int r = 0; r < 8; ++r) {
            int mr = r + ((lane >> 4) << 3);   // C/D layout: VGPR r -> M=r / M=r+8
            float val = (acc[r] + sOp[ntile * 256 + r * 32 + lane]) * sLinv[mr];
            out[(size_t)(b * Nsz + i0 + mr) * 64 + n] = val;
        }
    }
}

extern "C" void kernel_launch(void* const* d_in, const int* in_sizes, int n_in,
                              void* d_out, int out_size, void* d_ws, size_t ws_size,
                              hipStream_t stream) {
    (void)in_sizes; (void)n_in; (void)out_size; (void)ws_size;
    const float* qs   = (const float*)d_in[0];
    const float* ks   = (const float*)d_in[1];
    const float* vs   = (const float*)d_in[2];
    const float* W    = (const float*)d_in[3];
    const float* bias = (const float*)d_in[4];
    const float* a    = (const float*)d_in[5];
    float* out = (float*)d_out;

    float* Hq  = (float*)d_ws;            // 4096 x 64
    float* Hkb = Hq + 4096 * 64;          // 4096 x 64  (2 MB total workspace)

    proj_kernel<<<1024, 256, 0, stream>>>(qs, ks, W, bias, Hq, Hkb);
    attn_kernel<<<Bsz * (Nsz / 16), 256, 0, stream>>>(Hq, Hkb, vs, a, out);
}